// GENConv_27049704030783
// MI455X (gfx1250) — compile-verified
//
#include <hip/hip_runtime.h>

#define N_NODES 2048
#define N_EDGES 16384
#define BDIM 8
#define CDIM 64
#define TDIM 12
#define OUTD 64
#define BT (BDIM * TDIM) /* 96 */
#define EPSC 1e-7f

typedef __attribute__((ext_vector_type(2))) float v2f;
typedef __attribute__((ext_vector_type(8))) float v8f;

// ---------------- zero scratch ----------------
__global__ void zero_f4(float4* __restrict__ p, int n4) {
  int i = blockIdx.x * blockDim.x + threadIdx.x;
  int stride = gridDim.x * blockDim.x;
  float4 z = make_float4(0.f, 0.f, 0.f, 0.f);
  for (; i < n4; i += stride) p[i] = z;
}

// ---------------- edge pass: accumulate sum(e^m) and sum(m*e^m) per dst ----
// node_feats layout [B,C,T,N]; accumulators layout [B,T,C,N]
__global__ void edge_pass(const float* __restrict__ nf, const float* __restrict__ ef,
                          const int* __restrict__ src, const int* __restrict__ dst,
                          float* __restrict__ sumex, float* __restrict__ summex) {
  int e  = blockIdx.x * 32 + threadIdx.x;   // wave lane = edge
  int c  = blockIdx.y * 8 + threadIdx.y;    // channel fixed per wave
  int bt = blockIdx.z;
  int b = bt / TDIM, t = bt % TDIM;
  int s = src[e];
  int d = dst[e];
  // gather within the contiguous 8KB n-row for (b,c,t): L2-resident
  float x = nf[(((size_t)b * CDIM + c) * TDIM + t) * N_NODES + s] + ef[(size_t)e * CDIM + c];
  float m = fmaxf(x, 0.f) + EPSC;
  float w = __expf(m);  // v_exp_f32; shift-free softmax is exact (m bounded)
  size_t ob = ((size_t)bt * CDIM + c) * N_NODES + d;
  atomicAdd(&sumex[ob], w);
  atomicAdd(&summex[ob], m * w);
}

// ---------------- fused feats + GEMM + bias + transpose-store ---------------
__global__ __launch_bounds__(128) void fused_gemm(
    const float* __restrict__ nf, const float* __restrict__ sumex,
    const float* __restrict__ summex, const float* __restrict__ W,
    const float* __restrict__ bias, float* __restrict__ out) {
  __shared__ __align__(16) float sA[16 * 66];      // feats tile [n_local][c], pad 66
  __shared__ __align__(16) float sT[4][16 * 20];   // per-wave out tile [o][n], pad 20

  int bt = blockIdx.y;
  int b = bt / TDIM, t = bt % TDIM;
  int n0 = blockIdx.x * 16;
  int tid = threadIdx.x;
  int nl = tid & 15;        // node within tile
  int c0 = tid >> 4;        // 0..7

  // Phase 1: feats = h + (sumex>0 ? summex/sumex : 0) into LDS, all 128 threads
#pragma unroll
  for (int cc = 0; cc < 8; ++cc) {
    int c = c0 + cc * 8;
    float h = nf[(((size_t)b * CDIM + c) * TDIM + t) * N_NODES + n0 + nl];
    size_t off = ((size_t)bt * CDIM + c) * N_NODES + n0 + nl;
    float sx = sumex[off];
    float agg = sx > 0.f ? summex[off] / sx : 0.f;
    sA[nl * 66 + c] = h + agg;
  }
  __syncthreads();

  int wave = tid >> 5;
  int lane = tid & 31;
  int half = lane >> 4;     // lane-half selects K pair (ISA A/B layout)
  int l16 = lane & 15;
  int o0 = wave * 16;       // each wave owns one 16-wide output column tile

  // Preload B fragments: VGPR j, half h -> K = k0 + j + 2h ; N = l16
  v2f Bf[16];
#pragma unroll
  for (int kk = 0; kk < 16; ++kk) {
    int k = kk * 4 + 2 * half;
    Bf[kk].x = W[(size_t)k * OUTD + o0 + l16];
    Bf[kk].y = W[(size_t)(k + 1) * OUTD + o0 + l16];
  }

  v8f acc = {};
#pragma unroll
  for (int kk = 0; kk < 16; ++kk) {
    // A fragment: lane l16 holds rows, (v0,v1) = K = k0+2h, k0+2h+1 (b64 LDS read)
    v2f av = *(const v2f*)&sA[l16 * 66 + kk * 4 + 2 * half];
    acc = __builtin_amdgcn_wmma_f32_16x16x4_f32(
        false, av, false, Bf[kk], (short)0, acc, false, false);
  }

  // bias depends only on output column = l16 lane position
  float bb = bias[o0 + l16];
  // C/D layout: VGPR r -> n_local = r + 8*half, o_local = l16
#pragma unroll
  for (int r = 0; r < 8; ++r) {
    sT[wave][l16 * 20 + half * 8 + r] = acc[r] + bb;
  }
  __syncthreads();

  // Read back transposed: lane -> (o = lane/2, n half = lane&1), 2x float4
  int o = lane >> 1, nh = lane & 1;
  const float* sp = &sT[wave][o * 20 + nh * 8];
  float4 v0 = *(const float4*)(sp);
  float4 v1 = *(const float4*)(sp + 4);
  size_t oaddr = (((size_t)b * OUTD + o0 + o) * TDIM + t) * N_NODES + n0 + nh * 8;
  *(float4*)&out[oaddr] = v0;
  *(float4*)&out[oaddr + 4] = v1;
}

extern "C" void kernel_launch(void* const* d_in, const int* in_sizes, int n_in,
                              void* d_out, int out_size, void* d_ws, size_t ws_size,
                              hipStream_t stream) {
  const float* nf   = (const float*)d_in[0];  // [B,C,T,N]
  const float* ef   = (const float*)d_in[1];  // [E,1,1,C]
  const int*   src  = (const int*)d_in[2];
  const int*   dst  = (const int*)d_in[3];
  const float* W    = (const float*)d_in[4];  // [C,OUT]
  const float* bias = (const float*)d_in[5];  // [OUT]
  float* out = (float*)d_out;                 // [B,OUT,T,N]

  const size_t SZ = (size_t)BT * CDIM * N_NODES;  // 12,582,912 floats per buffer
  float* sumex  = (float*)d_ws;
  float* summex = sumex + SZ;

  int n4 = (int)((2 * SZ) / 4);
  zero_f4<<<2048, 256, 0, stream>>>((float4*)d_ws, n4);

  dim3 eb(32, 8, 1);
  dim3 eg(N_EDGES / 32, CDIM / 8, BT);
  edge_pass<<<eg, eb, 0, stream>>>(nf, ef, src, dst, sumex, summex);

  dim3 gg(N_NODES / 16, BT, 1);
  fused_gemm<<<gg, 128, 0, stream>>>(nf, sumex, summex, W, bias, out);
}